// DecoderBlock_10230612099622
// MI455X (gfx1250) — compile-verified
//
#include <hip/hip_runtime.h>
#include <hip/hip_bf16.h>

// ---------------------------------------------------------------------------
// SSM decoder block for MI455X (gfx1250, wave32, WMMA).
// Pipeline: discretize -> [chunk-ends scan] -> [carry combine] ->
//           fused v-WMMA + scan + y-WMMA + LN + SiLU -> upsample-conv WMMA.
// All GEMMs use V_WMMA_F32_16X16X4_F32 (full f32 precision; problem is
// HBM-bandwidth bound so the f32 WMMA rate is ample).
// Round-3 fixes:
//  * anti-hoisting barrier moved from the pointer to an SGPR *offset* so
//    addrspace inference still sees a global base -> global_load (round 2's
//    pointer-opaque produced flat_load, coupling LOADcnt with DScnt).
//  * prefetch next u subtile via __builtin_prefetch (global_prefetch_b8) to
//    hide HBM latency behind WMMA + scan.
// ---------------------------------------------------------------------------

#define BSZ   16
#define CIN   64
#define CO    64
#define TLEN  8192
#define NST   256
#define NCH   16          // time chunks
#define LCH   512         // chunk length
#define SUB   32          // timesteps per LDS subtile
#define NSUB  (LCH/SUB)   // 16 subtiles per chunk
#define T2    (TLEN/2)

#define UST   40          // lds_u row stride (dwords), c-major [CIN][SUB]
#define VST   260         // lds_v row stride, t-major [SUB][NST]
#define YST   66          // lds_y row stride, t-major [SUB][CO]

typedef __attribute__((ext_vector_type(2))) float v2f;
typedef __attribute__((ext_vector_type(8))) float v8f;

// D(16x16) = A(16x4) * B(4x16) + C   (f32 WMMA, 8-arg VOP3P form)
__device__ __forceinline__ v8f wmma4(v2f a, v2f b, v8f c) {
  return __builtin_amdgcn_wmma_f32_16x16x4_f32(
      /*neg_a=*/false, a, /*neg_b=*/false, b,
      /*c_mod=*/(short)0, c, /*reuse_a=*/false, /*reuse_b=*/false);
}

// Returns 0 in an SGPR that the optimizer cannot constant-fold/CSE.
// Adding it to a pointer defeats cross-iteration load hoisting while keeping
// the pointer's global address space visible (=> global_load, not flat_load).
__device__ __forceinline__ size_t opaque_zero() {
  size_t z = 0;
  asm volatile("" : "+s"(z));
  return z;
}

// ---------------------------------------------------------------------------
// Kernel 0: ZOH discretization. Ad[n] = exp(-softplus(rl[n])),
// Bd[c,n] = B_c[c,n] * (Ad-1)/lam  (DT=1).
// ---------------------------------------------------------------------------
__global__ void k_discretize(const float* __restrict__ rl,
                             const float* __restrict__ Bc,
                             float* __restrict__ Ad, float* __restrict__ Bd) {
  int n = threadIdx.x;            // 256 threads, 1 block
  float x = rl[n];
  float sp = fmaxf(x, 0.0f) + log1pf(expf(-fabsf(x)));   // stable softplus
  float lam = -sp;
  float ad = expf(lam);
  float factor = (fabsf(lam) > 1e-6f) ? (ad - 1.0f) / lam : 1.0f;
  Ad[n] = ad;
  for (int c = 0; c < CIN; ++c)
    Bd[c * NST + n] = Bc[c * NST + n] * factor;
}

// ---------------------------------------------------------------------------
// Shared tile helpers (256-thread workgroups, 8 waves)
// ---------------------------------------------------------------------------
__device__ __forceinline__ void load_u_tile(const float* __restrict__ u, int b,
                                            int t0, float* lds_u) {
  int tid = threadIdx.x;
#pragma unroll
  for (int kk = 0; kk < (CIN * SUB) / 256; ++kk) {
    int i = tid + kk * 256;
    int c = i >> 5, tl = i & 31;
    lds_u[c * UST + tl] = u[((size_t)(b * CIN + c)) * TLEN + t0 + tl];
  }
}

// Prefetch the next subtile's u tile (global_prefetch_b8; no LOADcnt cost).
__device__ __forceinline__ void prefetch_u_tile(const float* __restrict__ u,
                                                int b, int t0) {
  int tid = threadIdx.x;
#pragma unroll
  for (int kk = 0; kk < (CIN * SUB) / 256; ++kk) {
    int i = tid + kk * 256;
    int c = i >> 5, tl = i & 31;
    __builtin_prefetch(&u[((size_t)(b * CIN + c)) * TLEN + t0 + tl], 0, 1);
  }
}

// v(SUB x NST) = u_tile^T(SUB x CIN) * Bd(CIN x NST), result into lds_v.
// 2 M-tiles x 16 N-tiles = 32 tiles; 4 per wave.
__device__ __forceinline__ void v_tile_wmma(const float* lds_u,
                                            const float* __restrict__ Bd,
                                            float* lds_v) {
  int tid = threadIdx.x;
  int wave = tid >> 5, lane = tid & 31, lo = lane & 15, hi = lane >> 4;
#pragma unroll
  for (int idx = 0; idx < 4; ++idx) {
    int tile = wave * 4 + idx;
    int nt = tile >> 1, mt = tile & 1;
    int m = mt * 16 + lo, nn = nt * 16 + lo;
    v8f acc = {0.f, 0.f, 0.f, 0.f, 0.f, 0.f, 0.f, 0.f};
#pragma unroll 8
    for (int ks = 0; ks < CIN / 4; ++ks) {
      int k = ks * 4 + 2 * hi;
      v2f a;  a.x = lds_u[k * UST + m];       a.y = lds_u[(k + 1) * UST + m];
      v2f bb; bb.x = Bd[k * NST + nn];        bb.y = Bd[(k + 1) * NST + nn];
      acc = wmma4(a, bb, acc);
    }
#pragma unroll
    for (int r = 0; r < 8; ++r)
      lds_v[(mt * 16 + r + 8 * hi) * VST + nn] = acc[r];
  }
}

// ---------------------------------------------------------------------------
// Pass 1: per-chunk end state with zero initial state (for carry combine).
// grid = (chunk, b), 256 threads. Recomputes v via WMMA; state in registers.
// ---------------------------------------------------------------------------
__global__ void k_chunk_ends(const float* __restrict__ u,
                             const float* __restrict__ Ad,
                             const float* __restrict__ Bd_in,
                             float* __restrict__ ce) {
  __shared__ float lds_u[CIN * UST];
  __shared__ float lds_v[SUB * VST];
  int chunk = blockIdx.x, b = blockIdx.y, tid = threadIdx.x;
  float a_n = Ad[tid];
  float s = 0.0f;
  for (int st = 0; st < NSUB; ++st) {
    const float* Bd = Bd_in + opaque_zero();  // defeat cross-iter hoisting
    int t0 = chunk * LCH + st * SUB;
    load_u_tile(u, b, t0, lds_u);
    if (st + 1 < NSUB) prefetch_u_tile(u, b, t0 + SUB);
    __syncthreads();
    v_tile_wmma(lds_u, Bd, lds_v);
    __syncthreads();
#pragma unroll 8
    for (int tl = 0; tl < SUB; ++tl)
      s = s * a_n + lds_v[tl * VST + tid];
    __syncthreads();
  }
  ce[chunk * (BSZ * NST) + b * NST + tid] = s;
}

// ---------------------------------------------------------------------------
// Pass 2: sequential carry combine over 16 chunks per (b,n).
// carry[ch] = state entering chunk ch.  E_ch = a^512 * E_{ch-1} + e_ch.
// ---------------------------------------------------------------------------
__global__ void k_combine(const float* __restrict__ Ad,
                          const float* __restrict__ ce,
                          float* __restrict__ carry) {
  int n = threadIdx.x, idx = blockIdx.x * NST + n;
  float aL = Ad[n];
#pragma unroll
  for (int i = 0; i < 9; ++i) aL = aL * aL;   // a^512
  float s = 0.0f;
  for (int ch = 0; ch < NCH; ++ch) {
    carry[ch * (BSZ * NST) + idx] = s;
    s = aL * s + ce[ch * (BSZ * NST) + idx];
  }
}

// ---------------------------------------------------------------------------
// Pass 3 (main fused kernel): v-WMMA -> scan (in LDS, in-place) ->
// y = S*C via WMMA -> LayerNorm + SiLU -> yact[b,o,t].
// grid = (chunk, b), 256 threads = 8 waves.
// ---------------------------------------------------------------------------
__global__ void k_scan_y(const float* __restrict__ u,
                         const float* __restrict__ Ad,
                         const float* __restrict__ Bd_in,
                         const float* __restrict__ Cm_in,
                         const float* __restrict__ ln_w,
                         const float* __restrict__ ln_b,
                         const float* __restrict__ carry,
                         float* __restrict__ yact) {
  __shared__ float lds_u[CIN * UST];
  __shared__ float lds_v[SUB * VST];
  __shared__ float lds_y[SUB * YST];
  __shared__ float lds_red[SUB * 16];
  __shared__ float lds_mu[SUB];
  __shared__ float lds_rs[SUB];

  int chunk = blockIdx.x, b = blockIdx.y, tid = threadIdx.x;
  int wave = tid >> 5, lane = tid & 31, lo = lane & 15, hi = lane >> 4;
  float a_n = Ad[tid];
  float s = carry[chunk * (BSZ * NST) + b * NST + tid];

  for (int st = 0; st < NSUB; ++st) {
    const float* Bd = Bd_in + opaque_zero();  // defeat cross-iter hoisting
    const float* Cm = Cm_in + opaque_zero();
    int t0 = chunk * LCH + st * SUB;
    load_u_tile(u, b, t0, lds_u);
    if (st + 1 < NSUB) prefetch_u_tile(u, b, t0 + SUB);
    __syncthreads();
    v_tile_wmma(lds_u, Bd, lds_v);
    __syncthreads();

    // ---- scan: thread tid owns state column n=tid, in-place v -> S ----
#pragma unroll 8
    for (int tl = 0; tl < SUB; ++tl) {
      s = s * a_n + lds_v[tl * VST + tid];
      lds_v[tl * VST + tid] = s;
    }
    __syncthreads();

    // ---- y(SUB x 64) = S(SUB x 256) * C(256 x 64): 2x4 tiles, 1/wave ----
    {
      int mt = wave >> 2, nt = wave & 3;
      int m = mt * 16 + lo, oo = nt * 16 + lo;
      v8f acc = {0.f, 0.f, 0.f, 0.f, 0.f, 0.f, 0.f, 0.f};
#pragma unroll 8
      for (int ks = 0; ks < NST / 4; ++ks) {
        int k = ks * 4 + 2 * hi;
        v2f a;  a.x = lds_v[m * VST + k];  a.y = lds_v[m * VST + k + 1];
        v2f bb; bb.x = Cm[k * CO + oo];    bb.y = Cm[(k + 1) * CO + oo];
        acc = wmma4(a, bb, acc);
      }
#pragma unroll
      for (int r = 0; r < 8; ++r)
        lds_y[(mt * 16 + r + 8 * hi) * YST + oo] = acc[r];
    }
    __syncthreads();

    // ---- LayerNorm stats over 64 channels per t-row ----
    {
      int t = tid & 31, q = tid >> 5;   // 8 partial-reducers per row
      float sm = 0.f, s2 = 0.f;
#pragma unroll
      for (int j = 0; j < 8; ++j) {
        float v = lds_y[t * YST + q * 8 + j];
        sm += v; s2 += v * v;
      }
      lds_red[(t * 8 + q) * 2 + 0] = sm;
      lds_red[(t * 8 + q) * 2 + 1] = s2;
    }
    __syncthreads();
    if (tid < SUB) {
      float sm = 0.f, s2 = 0.f;
#pragma unroll
      for (int q = 0; q < 8; ++q) {
        sm += lds_red[(tid * 8 + q) * 2 + 0];
        s2 += lds_red[(tid * 8 + q) * 2 + 1];
      }
      float mu = sm * (1.0f / CO);
      float var = s2 * (1.0f / CO) - mu * mu;
      lds_mu[tid] = mu;
      lds_rs[tid] = rsqrtf(var + 1e-5f);
    }
    __syncthreads();

    // ---- normalize + SiLU, write yact[b,o,t] (lane-contiguous t) ----
    {
      int t = lane;
      float mu = lds_mu[t], rs = lds_rs[t];
#pragma unroll
      for (int j = 0; j < 8; ++j) {
        int o = wave * 8 + j;
        float v = (lds_y[t * YST + o] - mu) * rs * ln_w[o] + ln_b[o];
        v = v / (1.0f + __expf(-v));           // SiLU
        yact[((size_t)(b * CO + o)) * TLEN + t0 + t] = v;
      }
    }
    __syncthreads();
  }
}

// ---------------------------------------------------------------------------
// Pass 4: upsample-conv. out[b,o,2τ] = out[b,o,2τ+1] =
//   conv_b[o] + Σ_c w[o,2c]·yact[b,c,τ] + w[o,2c+1]·yact[b,c,τ+T/2].
// Two 64x64 f32 WMMA GEMMs over T/2 columns, duplicated float2 stores.
// grid = (T2/64, b), 128 threads = 4 waves, one τ-tile (16 cols) per wave.
// ---------------------------------------------------------------------------
__global__ void k_conv(const float* __restrict__ yact,
                       const float* __restrict__ conv_w,
                       const float* __restrict__ conv_b,
                       float* __restrict__ out) {
  int b = blockIdx.y, tid = threadIdx.x;
  int wave = tid >> 5, lane = tid & 31, lo = lane & 15, hi = lane >> 4;
  int tau0 = (blockIdx.x * 4 + wave) * 16;

  v8f acc[4];
#pragma unroll
  for (int mt = 0; mt < 4; ++mt)
#pragma unroll
    for (int r = 0; r < 8; ++r)
      acc[mt][r] = conv_b[mt * 16 + r + 8 * hi];   // fold bias into C

#pragma unroll 2
  for (int ks = 0; ks < CO / 4; ++ks) {
    int k = ks * 4 + 2 * hi;
    v2f be, bo;
    be.x = yact[((size_t)(b * CO + k)) * TLEN + tau0 + lo];
    be.y = yact[((size_t)(b * CO + k + 1)) * TLEN + tau0 + lo];
    bo.x = yact[((size_t)(b * CO + k)) * TLEN + T2 + tau0 + lo];
    bo.y = yact[((size_t)(b * CO + k + 1)) * TLEN + T2 + tau0 + lo];
#pragma unroll
    for (int mt = 0; mt < 4; ++mt) {
      int m = mt * 16 + lo;
      v2f ae, ao;
      ae.x = conv_w[m * 128 + 2 * k];
      ae.y = conv_w[m * 128 + 2 * (k + 1)];
      ao.x = conv_w[m * 128 + 2 * k + 1];
      ao.y = conv_w[m * 128 + 2 * (k + 1) + 1];
      acc[mt] = wmma4(ae, be, acc[mt]);
      acc[mt] = wmma4(ao, bo, acc[mt]);
    }
  }

#pragma unroll
  for (int mt = 0; mt < 4; ++mt)
#pragma unroll
    for (int r = 0; r < 8; ++r) {
      int o = mt * 16 + r + 8 * hi;
      int t = 2 * (tau0 + lo);
      float v = acc[mt][r];
      float2 pv = make_float2(v, v);
      *reinterpret_cast<float2*>(&out[((size_t)(b * CO + o)) * TLEN + t]) = pv;
    }
}

// ---------------------------------------------------------------------------
extern "C" void kernel_launch(void* const* d_in, const int* in_sizes, int n_in,
                              void* d_out, int out_size, void* d_ws, size_t ws_size,
                              hipStream_t stream) {
  const float* u     = (const float*)d_in[0];
  const float* rl    = (const float*)d_in[1];
  const float* Bc    = (const float*)d_in[2];
  const float* Cm    = (const float*)d_in[3];
  const float* ln_w  = (const float*)d_in[4];
  const float* ln_b  = (const float*)d_in[5];
  const float* cw    = (const float*)d_in[6];
  const float* cb    = (const float*)d_in[7];
  float* out = (float*)d_out;

  float* ws    = (float*)d_ws;
  float* Ad    = ws;                       // 256
  float* Bd    = Ad + NST;                 // 64*256
  float* ce    = Bd + CIN * NST;           // 16*4096
  float* carry = ce + NCH * BSZ * NST;     // 16*4096
  float* yact  = carry + NCH * BSZ * NST;  // 16*64*8192  (~33.5 MB total ws)

  k_discretize<<<1, NST, 0, stream>>>(rl, Bc, Ad, Bd);
  k_chunk_ends<<<dim3(NCH, BSZ), 256, 0, stream>>>(u, Ad, Bd, ce);
  k_combine<<<BSZ, NST, 0, stream>>>(Ad, ce, carry);
  k_scan_y<<<dim3(NCH, BSZ), 256, 0, stream>>>(u, Ad, Bd, Cm, ln_w, ln_b,
                                               carry, yact);
  k_conv<<<dim3(T2 / 64, BSZ), 128, 0, stream>>>(yact, cw, cb, out);
}